// Pointer_67362267070642
// MI455X (gfx1250) — compile-verified
//
#include <hip/hip_runtime.h>
#include <hip/hip_bf16.h>
#include <math.h>

// Problem sizes (fixed by the reference)
#define Bn 64
#define Sn 1024
#define Dn 256
#define Hn 256
#define NWG 256     // persistent workgroups (= #16x16 tiles of the gates matrix)
#define TPB 256

typedef float v2f __attribute__((ext_vector_type(2)));
typedef float v8f __attribute__((ext_vector_type(8)));
typedef int   v4i __attribute__((ext_vector_type(4)));

#define AS1 __attribute__((address_space(1)))
#define AS3 __attribute__((address_space(3)))

// ---- compile-time probes (diagnostics only; never fail the build) ----------
#if !__has_builtin(__builtin_amdgcn_tanhf)
#warning "probe: no __builtin_amdgcn_tanhf (falling back to ocml tanhf)"
#endif
#if __has_builtin(__builtin_amdgcn_global_load_async_to_lds_b128) && \
    __has_builtin(__builtin_amdgcn_s_wait_asynccnt)
#define USE_ASYNC_LDS 1
#else
#warning "probe: async-to-LDS builtins unavailable (using sync LDS fill)"
#endif
#if !__has_builtin(__builtin_amdgcn_tensor_load_to_lds)
#warning "probe: no __builtin_amdgcn_tensor_load_to_lds"
#endif

static __device__ __forceinline__ float fast_tanh(float x){
#if __has_builtin(__builtin_amdgcn_tanhf)
  return __builtin_amdgcn_tanhf(x);
#else
  return tanhf(x);
#endif
}
static __device__ __forceinline__ float sigm(float x){
  return 1.0f / (1.0f + __expf(-x));
}

// ---------------------------------------------------------------------------
// key_out[m,n] = enc[m,:] . W1[n,:] + b1[n]    (fp32 WMMA, 16x64 per wave)
// ---------------------------------------------------------------------------
__global__ void k_keyproj(const float* __restrict__ enc, const float* __restrict__ W1,
                          const float* __restrict__ b1, float* __restrict__ key)
{
  const int wave = threadIdx.x >> 5, lane = threadIdx.x & 31;
  const int half = lane >> 4, l16 = lane & 15;
  const int nbase = blockIdx.x * 64;
  const int m0 = (blockIdx.y * 4 + wave) * 16;

  const float* arow = enc + (size_t)(m0 + l16) * Hn + 2 * half;
  const float* br0 = W1 + (size_t)(nbase +  0 + l16) * Hn + 2 * half;
  const float* br1 = W1 + (size_t)(nbase + 16 + l16) * Hn + 2 * half;
  const float* br2 = W1 + (size_t)(nbase + 32 + l16) * Hn + 2 * half;
  const float* br3 = W1 + (size_t)(nbase + 48 + l16) * Hn + 2 * half;

  v8f a0 = {0,0,0,0,0,0,0,0}, a1 = a0, a2 = a0, a3 = a0;
  #pragma unroll 4
  for (int k = 0; k < Hn; k += 4) {
    v2f a = *(const v2f*)(arow + k);
    a0 = __builtin_amdgcn_wmma_f32_16x16x4_f32(false, a, false, *(const v2f*)(br0 + k), (short)0, a0, false, false);
    a1 = __builtin_amdgcn_wmma_f32_16x16x4_f32(false, a, false, *(const v2f*)(br1 + k), (short)0, a1, false, false);
    a2 = __builtin_amdgcn_wmma_f32_16x16x4_f32(false, a, false, *(const v2f*)(br2 + k), (short)0, a2, false, false);
    a3 = __builtin_amdgcn_wmma_f32_16x16x4_f32(false, a, false, *(const v2f*)(br3 + k), (short)0, a3, false, false);
  }
  #pragma unroll
  for (int i = 0; i < 4; ++i) {
    v8f acc = (i == 0) ? a0 : (i == 1) ? a1 : (i == 2) ? a2 : a3;
    int n = nbase + i * 16 + l16;
    float bias = b1[n];
    #pragma unroll
    for (int j = 0; j < 8; ++j) {
      int row = m0 + j + 8 * half;
      key[(size_t)row * Hn + n] = acc[j] + bias;
    }
  }
}

// ---------------------------------------------------------------------------
// d[b,t,k] = sigmoid(x_shift . W_in^T + b_in), stored transposed:
//   DT[(t*H + k)*B + b]
// ---------------------------------------------------------------------------
__global__ void k_dproj(const float* __restrict__ x, const float* __restrict__ Win,
                        const float* __restrict__ bin, float* __restrict__ DT)
{
  const int wave = threadIdx.x >> 5, lane = threadIdx.x & 31;
  const int half = lane >> 4, l16 = lane & 15;
  const int nbase = blockIdx.x * 64;
  const int m0 = (blockIdx.y * 4 + wave) * 16;

  const int arowm = m0 + l16;
  const int trow  = arowm & (Sn - 1);
  const int brows = arowm >> 10;
  const int tprev = (trow == 0) ? 0 : (trow - 1);
  const float zmul = (trow == 0) ? 0.f : 1.f;

  const float* arow = x + ((size_t)brows * Sn + tprev) * Dn + 2 * half;
  const float* br0 = Win + (size_t)(nbase +  0 + l16) * Dn + 2 * half;
  const float* br1 = Win + (size_t)(nbase + 16 + l16) * Dn + 2 * half;
  const float* br2 = Win + (size_t)(nbase + 32 + l16) * Dn + 2 * half;
  const float* br3 = Win + (size_t)(nbase + 48 + l16) * Dn + 2 * half;

  v8f a0 = {0,0,0,0,0,0,0,0}, a1 = a0, a2 = a0, a3 = a0;
  #pragma unroll 4
  for (int k = 0; k < Dn; k += 4) {
    v2f a = *(const v2f*)(arow + k);
    a.x *= zmul; a.y *= zmul;
    a0 = __builtin_amdgcn_wmma_f32_16x16x4_f32(false, a, false, *(const v2f*)(br0 + k), (short)0, a0, false, false);
    a1 = __builtin_amdgcn_wmma_f32_16x16x4_f32(false, a, false, *(const v2f*)(br1 + k), (short)0, a1, false, false);
    a2 = __builtin_amdgcn_wmma_f32_16x16x4_f32(false, a, false, *(const v2f*)(br2 + k), (short)0, a2, false, false);
    a3 = __builtin_amdgcn_wmma_f32_16x16x4_f32(false, a, false, *(const v2f*)(br3 + k), (short)0, a3, false, false);
  }
  #pragma unroll
  for (int i = 0; i < 4; ++i) {
    v8f acc = (i == 0) ? a0 : (i == 1) ? a1 : (i == 2) ? a2 : a3;
    int n = nbase + i * 16 + l16;
    float bias = bin[n];
    #pragma unroll
    for (int j = 0; j < 8; ++j) {
      int row = m0 + j + 8 * half;
      int tt = row & (Sn - 1);
      int bb = row >> 10;
      DT[((size_t)tt * Hn + n) * Bn + bb] = sigm(acc[j] + bias);
    }
  }
}

// ---------------------------------------------------------------------------
__global__ void k_init(const float* __restrict__ h0, float* __restrict__ hT,
                       unsigned* __restrict__ bar, unsigned long long* __restrict__ packed)
{
  int i = blockIdx.x * blockDim.x + threadIdx.x;
  if (i < Bn * Hn) {
    int b = i / Hn, j = i % Hn;
    hT[j * Bn + b] = h0[i];
  }
  if (i < Bn * Sn) packed[i] = 0ull;
  if (i == 0) *bar = 0u;
}

__global__ void k_final(const unsigned long long* __restrict__ packed, int* __restrict__ preds)
{
  int i = blockIdx.x * blockDim.x + threadIdx.x;
  if (i < Bn * Sn) preds[i] = (int)(~(unsigned)(packed[i] & 0xffffffffull));
}

static __device__ __forceinline__ void grid_barrier(unsigned* bar, unsigned target)
{
  __syncthreads();
  if (threadIdx.x == 0) {
    __threadfence();
    atomicAdd(bar, 1u);
    while (__hip_atomic_load(bar, __ATOMIC_RELAXED, __HIP_MEMORY_SCOPE_AGENT) < target)
      __builtin_amdgcn_s_sleep(2);
    __threadfence();
  }
  __syncthreads();
}

// ---------------------------------------------------------------------------
// Persistent scan: 256 WGs x 256 threads, 1024 steps.
//  Phase A : one 16x16 WMMA gates tile per WG (n = j*4+g interleaved);
//            d-half of the A panel staged in LDS via async DMA
//            (global_load_async_to_lds_b128, ASYNCcnt), refilled one step
//            AHEAD so the copy overlaps barriers and phases B/C.
//  Phase B : q = h_new @ W2^T + b2 (WGs 0..63, scalar, tiny).
//  Phase C : attention row sweep + fused argmax via packed atomicMax_u64.
// ---------------------------------------------------------------------------
__global__ void __launch_bounds__(TPB, 1) k_scan(
    const float* __restrict__ DT,  const float* __restrict__ key,
    const float* __restrict__ Wih, const float* __restrict__ bih,
    const float* __restrict__ Whh, const float* __restrict__ bhh,
    const float* __restrict__ W2,  const float* __restrict__ b2,
    const float* __restrict__ vw,  const float* __restrict__ vb,
    const unsigned char* __restrict__ mask,
    const float* __restrict__ c0,
    float* __restrict__ hbuf, float* __restrict__ qbuf,
    unsigned* __restrict__ bar,
    float* __restrict__ logits, unsigned long long* __restrict__ packed)
{
  __shared__ float s_w[16][516];   // B operand rows (pad 516: conflict-free)
  __shared__ float s_a[256][20];   // d-panel [k][b-local] (pad 20: 16B rows)
  __shared__ float s_w2[4][Hn];
  __shared__ float s_red[16];

  const int w    = blockIdx.x;
  const int tid  = threadIdx.x;
  const int lane = tid & 31;
  const int wvid = tid >> 5;
  const int half = lane >> 4, l16 = lane & 15;

  const int mb = w >> 6;           // b-tile 0..3
  const int nb = w & 63;           // 4-j block 0..63

  // stage step-invariant B operand (rows: n = nb*16 + r, cols: [Wih | Whh])
  for (int idx = tid; idx < 16 * 512; idx += TPB) {
    int r = idx >> 9, k = idx & 511;
    int jg = nb * 4 + (r >> 2);
    int g  = r & 3;
    s_w[r][k] = (k < 256) ? Wih[(size_t)(g * Hn + jg) * Hn + k]
                          : Whh[(size_t)(g * Hn + jg) * Hn + (k - 256)];
  }
  if (w < 64) {
    for (int idx = tid; idx < 4 * Hn; idx += TPB) {
      int r = idx >> 8, k = idx & 255;
      s_w2[r][k] = W2[(size_t)(w * 4 + r) * Hn + k];
    }
  }

  // cooperative d-panel fill (async DMA when available)
  auto fill_panel = [&](int t2) {
    const float* src = DT + (size_t)t2 * Hn * Bn + mb * 16;
    for (int v = tid; v < 1024; v += TPB) {       // 1024 x float4 = 16 KB
      int k = v >> 2, bq = (v & 3) * 4;
      const float* g = src + (size_t)k * Bn + bq;
#ifdef USE_ASYNC_LDS
      __builtin_amdgcn_global_load_async_to_lds_b128(
          (AS1 v4i*)g, (AS3 v4i*)&s_a[k][bq], 0, 0);
#else
      float4 tmp = *(const float4*)g;
      *(float4*)&s_a[k][bq] = tmp;
#endif
    }
  };
  fill_panel(0);

  // per-lane constants for phase A (wave 0 only)
  const int jloc  = l16 >> 2;
  const int jglob = nb * 4 + jloc;
  float creg[8];
  float cbias = 0.f;
  if (wvid == 0) {
    int g = l16 & 3;
    cbias = bih[g * Hn + jglob] + bhh[g * Hn + jglob];
    #pragma unroll
    for (int jr = 0; jr < 8; ++jr) {
      int brow = mb * 16 + jr + 8 * half;
      creg[jr] = c0[brow * Hn + jglob];
    }
  }

  // per-lane constants for phases B/C
  const float qb = (w < 64) ? b2[w * 4 + (tid >> 6)] : 0.f;
  float vv[8];
  #pragma unroll
  for (int u = 0; u < 8; ++u) vv[u] = vw[lane * 8 + u];
  const float vbias = vb[0];
  const int bC = w >> 2, sq = w & 3;

  for (int t = 0; t < Sn; ++t) {
    const float* hcur = hbuf + (t & 1) * (Bn * Hn);
    float*       hnxt = hbuf + ((t + 1) & 1) * (Bn * Hn);

    // ---- Phase A ---------------------------------------------------------
#ifdef USE_ASYNC_LDS
    __builtin_amdgcn_s_wait_asynccnt(0);   // each wave's panel DMA complete
#endif
    __syncthreads();                       // panel visible WG-wide
    if (wvid == 0) {
      v8f acc = {0,0,0,0,0,0,0,0};
      #pragma unroll 4
      for (int kk = 0; kk < 256; kk += 4) {          // d-half from LDS
        int k0 = kk + 2 * half;
        v2f a = { s_a[k0][l16], s_a[k0 + 1][l16] };
        v2f bf = *(const v2f*)&s_w[l16][k0];
        acc = __builtin_amdgcn_wmma_f32_16x16x4_f32(false, a, false, bf, (short)0, acc, false, false);
      }
      const float* hptr = hcur + (mb * 16 + l16);
      #pragma unroll 4
      for (int kk = 0; kk < 256; kk += 4) {          // h-half from global
        int k0 = kk + 2 * half;
        v2f a = { hptr[(size_t)k0 * Bn], hptr[(size_t)(k0 + 1) * Bn] };
        v2f bf = *(const v2f*)&s_w[l16][256 + k0];
        acc = __builtin_amdgcn_wmma_f32_16x16x4_f32(false, a, false, bf, (short)0, acc, false, false);
      }
      float av[8];
      #pragma unroll
      for (int jr = 0; jr < 8; ++jr) av[jr] = acc[jr] + cbias;
      const int base = lane & ~3;
      #pragma unroll
      for (int jr = 0; jr < 8; ++jr) {
        float ig = __shfl(av[jr], base + 0, 32);
        float fg = __shfl(av[jr], base + 1, 32);
        float gg = __shfl(av[jr], base + 2, 32);
        float og = __shfl(av[jr], base + 3, 32);
        float cn = sigm(fg) * creg[jr] + sigm(ig) * fast_tanh(gg);
        float hn = sigm(og) * fast_tanh(cn);
        creg[jr] = cn;
        if ((lane & 3) == 0) {
          int brow = mb * 16 + jr + 8 * half;
          hnxt[jglob * Bn + brow] = hn;
        }
      }
    }
    __syncthreads();                       // wave0 done reading s_a
    if (t + 1 < Sn) fill_panel(t + 1);     // prefetch next d-panel (overlaps)
    grid_barrier(bar, (unsigned)(2 * t + 1) * NWG);

    // ---- Phase B ---------------------------------------------------------
    if (w < 64) {
      const int bB = tid & 63, jj = tid >> 6;
      const float* hp2 = hnxt + bB;
      float acc = qb;
      #pragma unroll 2
      for (int k = 0; k < Hn; ++k)
        acc += hp2[k * Bn] * s_w2[jj][k];
      qbuf[(t & 1) * (Bn * Hn) + bB * Hn + (w * 4 + jj)] = acc;
    }
    grid_barrier(bar, (unsigned)(2 * t + 2) * NWG);

    // ---- Phase C ---------------------------------------------------------
    {
      const float* q = qbuf + (t & 1) * (Bn * Hn) + bC * Hn;
      float qr[8];
      #pragma unroll
      for (int u = 0; u < 8; ++u) qr[u] = q[lane * 8 + u];

      const float* krow = key + (size_t)bC * Sn * Hn + lane * 8;
      float* lrow = logits + ((size_t)bC * Sn + t) * Sn;
      const unsigned char* mrow = mask + (size_t)bC * Sn;

      const int s_lo = sq * 256 + wvid * 32, s_hi = s_lo + 32;
      float bmax = -INFINITY; int bidx = s_lo;
      for (int s = s_lo; s < s_hi; ++s) {
        const float* kp = krow + (size_t)s * Hn;
        __builtin_prefetch(krow + (size_t)(s + 8) * Hn, 0, 1);
        float4 kA = *(const float4*)(kp);
        float4 kB = *(const float4*)(kp + 4);
        float a0 = fast_tanh(kA.x + qr[0]) * vv[0]
                 + fast_tanh(kA.y + qr[1]) * vv[1]
                 + fast_tanh(kA.z + qr[2]) * vv[2]
                 + fast_tanh(kA.w + qr[3]) * vv[3]
                 + fast_tanh(kB.x + qr[4]) * vv[4]
                 + fast_tanh(kB.y + qr[5]) * vv[5]
                 + fast_tanh(kB.z + qr[6]) * vv[6]
                 + fast_tanh(kB.w + qr[7]) * vv[7];
        #pragma unroll
        for (int off = 16; off; off >>= 1)
          a0 += __shfl_xor(a0, off, 32);
        float ui = a0 + vbias;
        if (!mrow[s]) ui = -INFINITY;
        if (lane == 0) lrow[s] = ui;
        if (ui > bmax) { bmax = ui; bidx = s; }
      }
      if (lane == 0) { s_red[wvid] = bmax; s_red[8 + wvid] = __int_as_float(bidx); }
      __syncthreads();
      if (tid == 0) {
        float mv = s_red[0]; int mi = __float_as_int(s_red[8]);
        #pragma unroll
        for (int u = 1; u < 8; ++u) {
          float v2 = s_red[u]; int i2 = __float_as_int(s_red[8 + u]);
          if (v2 > mv) { mv = v2; mi = i2; }
        }
        unsigned uo = __float_as_uint(mv);
        uo = (uo & 0x80000000u) ? ~uo : (uo | 0x80000000u);
        unsigned long long keyv = ((unsigned long long)uo << 32) | (unsigned)(~(unsigned)mi);
        atomicMax(&packed[(size_t)bC * Sn + t], keyv);
      }
      __syncthreads();
    }
  }
}

// ---------------------------------------------------------------------------
extern "C" void kernel_launch(void* const* d_in, const int* in_sizes, int n_in,
                              void* d_out, int out_size, void* d_ws, size_t ws_size,
                              hipStream_t stream)
{
  const float* x    = (const float*)d_in[0];
  const float* enc  = (const float*)d_in[1];
  const unsigned char* mask = (const unsigned char*)d_in[2];
  const float* h0   = (const float*)d_in[3];
  const float* c0   = (const float*)d_in[4];
  const float* Win  = (const float*)d_in[5];
  const float* bin  = (const float*)d_in[6];
  const float* Wih  = (const float*)d_in[7];
  const float* bih  = (const float*)d_in[8];
  const float* Whh  = (const float*)d_in[9];
  const float* bhh  = (const float*)d_in[10];
  const float* W1   = (const float*)d_in[11];
  const float* b1   = (const float*)d_in[12];
  const float* W2   = (const float*)d_in[13];
  const float* b2   = (const float*)d_in[14];
  const float* vw   = (const float*)d_in[15];
  const float* vb   = (const float*)d_in[16];

  char* ws = (char*)d_ws;
  float* keyb = (float*)ws; ws += (size_t)Bn * Sn * Hn * sizeof(float);   // 67 MB
  float* DT   = (float*)ws; ws += (size_t)Bn * Sn * Hn * sizeof(float);   // 67 MB
  float* hbuf = (float*)ws; ws += (size_t)2 * Bn * Hn * sizeof(float);
  float* qbuf = (float*)ws; ws += (size_t)2 * Bn * Hn * sizeof(float);
  unsigned long long* packed = (unsigned long long*)ws; ws += (size_t)Bn * Sn * sizeof(unsigned long long);
  unsigned* bar = (unsigned*)ws;

  float* logits = (float*)d_out;
  int*   preds  = (int*)((float*)d_out + (size_t)Bn * Sn * Sn);

  k_init<<<(Bn * Sn + 255) / 256, 256, 0, stream>>>(h0, hbuf, bar, packed);

  dim3 gproj(Hn / 64, (Bn * Sn) / 64);
  k_keyproj<<<gproj, 128, 0, stream>>>(enc, W1, b1, keyb);
  k_dproj  <<<gproj, 128, 0, stream>>>(x, Win, bin, DT);

  k_scan<<<NWG, TPB, 0, stream>>>(DT, keyb, Wih, bih, Whh, bhh, W2, b2,
                                  vw, vb, mask, c0, hbuf, qbuf, bar,
                                  logits, packed);

  k_final<<<(Bn * Sn + 255) / 256, 256, 0, stream>>>(packed, preds);
}